// GCN_84731114815818
// MI455X (gfx1250) — compile-verified
//
#include <hip/hip_runtime.h>

// ---------------------------------------------------------------------------
// GCN layer for MI455X (gfx1250, wave32):
//   summed[dst] += feature[src]  (f32 atomics, L2-resident: both tables ~26MB
//                                 each, global L2 is 192MB)
//   h = deg>0 ? summed/deg : feature
//   out = relu(h @ W + b)        (V_WMMA_F32_16X16X4_F32, fp32 end-to-end)
// ---------------------------------------------------------------------------

typedef __attribute__((ext_vector_type(2))) float v2f;
typedef __attribute__((ext_vector_type(8))) float v8f;

#define NFEAT 64

// ---- kernel 1: zero the accumulator workspace (summed + deg, contiguous) ---
__global__ __launch_bounds__(256) void gcn_zero(float* __restrict__ p, long n) {
    long stride = (long)gridDim.x * blockDim.x;
    long i = (long)blockIdx.x * blockDim.x + threadIdx.x;
    float4* p4 = (float4*)p;
    long n4 = n >> 2;
    for (long k = i; k < n4; k += stride)
        p4[k] = make_float4(0.f, 0.f, 0.f, 0.f);
    for (long k = (n4 << 2) + i; k < n; k += stride)
        p[k] = 0.f;
}

// ---- kernel 2: edge scatter. 16 threads per edge, float4 per thread --------
__global__ __launch_bounds__(256) void gcn_scatter(
    const float* __restrict__ feature,
    const long long* __restrict__ src,
    const long long* __restrict__ dst,
    float* __restrict__ summed,
    float* __restrict__ deg,
    long total /* = E*16 */) {
    long t = (long)blockIdx.x * blockDim.x + threadIdx.x;
    if (t >= total) return;
    int e = (int)(t >> 4);
    int g = (int)(t & 15);
    int s = (int)src[e];
    int d = (int)dst[e];
    // coalesced 128b gather of this edge's source row chunk (hits L2)
    const float4* frow = (const float4*)(feature + (long)s * NFEAT);
    float4 v = frow[g];
    float* o = summed + (long)d * NFEAT + g * 4;
    atomicAdd(o + 0, v.x);
    atomicAdd(o + 1, v.y);
    atomicAdd(o + 2, v.z);
    atomicAdd(o + 3, v.w);
    if (g == 0) atomicAdd(deg + d, 1.0f);
}

// ---- kernel 3: mean + fallback + relu(h@W + b) via f32 WMMA ----------------
// Block = 256 threads = 8 waves; each wave owns one 16-node tile.
__global__ __launch_bounds__(256) void gcn_gemm(
    const float* __restrict__ feature,
    const float* __restrict__ summed,
    const float* __restrict__ deg,
    const float* __restrict__ W,     // [64][64] row-major: W[k][n]
    const float* __restrict__ bias,  // [64]
    float* __restrict__ out,         // [N][64]
    int N, int nTiles) {
    __shared__ float Wl[NFEAT][NFEAT + 1];   // +1 pad: kill LDS bank conflicts
    __shared__ float bl[NFEAT];
    __shared__ float hl[8][16][NFEAT + 1];   // per-wave h tile

    const int tid = threadIdx.x;

    // Stage W and b into LDS (shared by all 8 waves).
    for (int i = tid; i < NFEAT * NFEAT; i += 256)
        Wl[i >> 6][i & 63] = W[i];
    if (tid < NFEAT) bl[tid] = bias[tid];
    __syncthreads();

    const int wave = tid >> 5;
    const int lane = tid & 31;
    const int tile = blockIdx.x * 8 + wave;
    if (tile >= nTiles) return;          // wave-uniform exit (no WMMA divergence)
    const int node0 = tile * 16;

    // Build h tile in LDS: 2 lanes per row, 32 cols each (8x float4).
    {
        const int r  = lane >> 1;
        const int cb = (lane & 1) * 32;
        const int node = node0 + r;
        if (node < N) {
            const float dg = deg[node];
            const float4* s4 = (const float4*)(summed + (long)node * NFEAT + cb);
            const float4* f4 = (const float4*)(feature + (long)node * NFEAT + cb);
            #pragma unroll
            for (int j = 0; j < 8; ++j) {
                float4 sv = s4[j];
                float4 fv = f4[j];
                float4 h;
                if (dg > 0.f) { h.x = sv.x / dg; h.y = sv.y / dg;
                                h.z = sv.z / dg; h.w = sv.w / dg; }
                else          { h = fv; }
                const int c = cb + j * 4;
                hl[wave][r][c + 0] = h.x;
                hl[wave][r][c + 1] = h.y;
                hl[wave][r][c + 2] = h.z;
                hl[wave][r][c + 3] = h.w;
            }
        } else {
            #pragma unroll
            for (int j = 0; j < 8; ++j) {
                const int c = cb + j * 4;
                hl[wave][r][c + 0] = 0.f; hl[wave][r][c + 1] = 0.f;
                hl[wave][r][c + 2] = 0.f; hl[wave][r][c + 3] = 0.f;
            }
        }
    }
    // Same-wave LDS RAW: compiler inserts s_wait_dscnt; no barrier needed
    // (h tile is private to this wave).

    // WMMA: A 16x4 layout -> lanes 0-15 hold K=k0,k0+1; lanes 16-31 K=k0+2,k0+3.
    // B 4x16 mirrored: N across lanes 0-15 / 16-31, K pair across the 2 VGPRs.
    const int mrow = lane & 15;
    const int khi  = (lane >> 4) * 2;

    v8f acc[4] = {};
    #pragma unroll
    for (int k0 = 0; k0 < NFEAT; k0 += 4) {
        const int kr = k0 + khi;
        v2f a;
        a.x = hl[wave][mrow][kr];
        a.y = hl[wave][mrow][kr + 1];
        #pragma unroll
        for (int nt = 0; nt < 4; ++nt) {
            v2f bf;
            bf.x = Wl[kr][nt * 16 + mrow];
            bf.y = Wl[kr + 1][nt * 16 + mrow];
            acc[nt] = __builtin_amdgcn_wmma_f32_16x16x4_f32(
                false, a, false, bf, (short)0, acc[nt], false, false);
        }
    }

    // Epilogue: C/D layout -> VGPR j holds rows j (lanes 0-15) / j+8 (16-31).
    const int rowOff = (lane >> 4) * 8;
    #pragma unroll
    for (int nt = 0; nt < 4; ++nt) {
        const int col = nt * 16 + mrow;
        const float bb = bl[col];
        #pragma unroll
        for (int j = 0; j < 8; ++j) {
            const int row = node0 + rowOff + j;
            if (row < N)
                out[(long)row * NFEAT + col] = fmaxf(acc[nt][j] + bb, 0.f);
        }
    }
}

// ---------------------------------------------------------------------------
extern "C" void kernel_launch(void* const* d_in, const int* in_sizes, int n_in,
                              void* d_out, int out_size, void* d_ws, size_t ws_size,
                              hipStream_t stream) {
    const float*      feature = (const float*)d_in[0];
    const long long*  src     = (const long long*)d_in[1];  // int64 indices
    const long long*  dst     = (const long long*)d_in[2];
    const float*      W       = (const float*)d_in[3];
    const float*      bias    = (const float*)d_in[4];
    float*            out     = (float*)d_out;

    const int N = in_sizes[0] / NFEAT;
    const int E = in_sizes[1];

    float* summed = (float*)d_ws;                 // N*64 f32
    float* deg    = summed + (size_t)N * NFEAT;   // N f32

    // 1) zero accumulators (N*65 floats, ~26 MB -> ~1 us of HBM writes)
    const long zn = (long)N * (NFEAT + 1);
    gcn_zero<<<2048, 256, 0, stream>>>(summed, zn);

    // 2) edge scatter (E*16 threads)
    const long total = (long)E * 16;
    const int sblocks = (int)((total + 255) / 256);
    gcn_scatter<<<sblocks, 256, 0, stream>>>(feature, src, dst, summed, deg, total);

    // 3) mean + dense layer via f32 WMMA
    const int nTiles = (N + 15) / 16;
    const int gblocks = (nTiles + 7) / 8;
    gcn_gemm<<<gblocks, 256, 0, stream>>>(feature, summed, deg, W, bias, out,
                                          N, nTiles);
}